// LightSpatialAttention3d_82944408420538
// MI455X (gfx1250) — compile-verified
//
#include <hip/hip_runtime.h>
#include <cstdint>

typedef __attribute__((ext_vector_type(2))) float v2f;
typedef __attribute__((ext_vector_type(4))) float v4f;
typedef __attribute__((ext_vector_type(8))) float v8f;

// Problem geometry (fixed by the reference)
constexpr int Bsz = 2, Csz = 32, Dsz = 64, Hsz = 128, Wsz = 128;
constexpr int HW  = Hsz * Wsz;          // 16384
constexpr int DHW = Dsz * HW;           // 1,048,576
constexpr int S_SPATIAL = Bsz * DHW;    // 2,097,152 (per pooled plane)

// Per-tap LDS row base (row stride 132, slot = wpos + kw), folds to immediates
// after unrolling. Taps >= 54 are zero-weight padding -> point at slot 0
// (always zeroed), product is 0 regardless.
__host__ __device__ constexpr int tapBase(int t) {
    if (t >= 54) return 0;
    int ci = t / 27, rr = t % 27;
    int zz = rr / 9, yy = (rr % 9) / 3, kw = rr % 3;
    return ((ci * 3 + zz) * 3 + yy) * 132 + kw;
}

// ---------------------------------------------------------------------------
// Kernel 1: fused channel max + mean, one NT pass over x (256 MB), b128 wide.
// Writes maxP[S_SPATIAL] | avgP[S_SPATIAL] into workspace (regular temporal:
// 16 MB, we want it resident in the 192 MB L2 for kernel 2).
// ---------------------------------------------------------------------------
__global__ __launch_bounds__(256) void pool_kernel(const float* __restrict__ x,
                                                   float* __restrict__ maxP,
                                                   float* __restrict__ avgP) {
    int g = blockIdx.x * 256 + threadIdx.x;           // [0, S_SPATIAL/4)
    long s   = (long)g * 4;
    long b   = s / DHW;
    long rem = s - b * DHW;
    const v4f* xb = (const v4f*)(x + b * (long)Csz * DHW + rem);

    v4f v = __builtin_nontemporal_load(xb);
    v4f mx = v, sm = v;
#pragma unroll
    for (int c = 1; c < Csz; ++c) {
        v4f t = __builtin_nontemporal_load(xb + (long)c * (DHW / 4));
        mx.x = fmaxf(mx.x, t.x); sm.x += t.x;
        mx.y = fmaxf(mx.y, t.y); sm.y += t.y;
        mx.z = fmaxf(mx.z, t.z); sm.z += t.z;
        mx.w = fmaxf(mx.w, t.w); sm.w += t.w;
    }
    const float inv = 1.0f / (float)Csz;
    v4f ao = sm * inv;
    ((v4f*)maxP)[g] = mx;
    ((v4f*)avgP)[g] = ao;
}

// ---------------------------------------------------------------------------
// Kernel 2: one block per (b,d,h) row of 128 outputs.
//   stage 18 pooled halo rows -> LDS via global_load_async_to_lds_b32
//   zero OOB slots (scalar row test), then 14 chained V_WMMA_F32_16X16X4_F32
//   per wave compute attn; finally NT b128 stream: out = x * sigmoid(attn).
// ---------------------------------------------------------------------------
__global__ __launch_bounds__(256) void attn_kernel(const float* __restrict__ x,
                                                   const float* __restrict__ cw,
                                                   const float* __restrict__ cb,
                                                   const float* __restrict__ pooled,
                                                   float* __restrict__ out) {
    __shared__ float tile[18 * 132];                // slot xx holds w = xx-1
    __shared__ __align__(16) float wLds[56];        // weights, K padded to 56
    __shared__ __align__(16) float sAttn[Wsz];

    const int bid = blockIdx.x;                     // [0, Bsz*Dsz*Hsz)
    const int b   = bid / (Dsz * Hsz);
    const int r   = bid - b * (Dsz * Hsz);
    const int d   = r / Hsz;
    const int h   = r - d * Hsz;
    const int tid = threadIdx.x;

    if (tid < 56) wLds[tid] = (tid < 54) ? cw[tid] : 0.0f;

    // ---- stage: 18 rows x 132 slots, clamped (full EXEC for async ops) ----
    const int xx  = tid < 132 ? tid : 131;          // dup lanes hit slot 131 (never read)
    int wcl = xx - 1; wcl = wcl < 0 ? 0 : (wcl > Wsz - 1 ? Wsz - 1 : wcl);
#pragma unroll
    for (int rr = 0; rr < 18; ++rr) {
        const int ci = rr / 9, zz = (rr / 3) % 3, yy = rr % 3;   // compile-time
        int z = d + zz - 1; z = z < 0 ? 0 : (z > Dsz - 1 ? Dsz - 1 : z);  // scalar
        int y = h + yy - 1; y = y < 0 ? 0 : (y > Hsz - 1 ? Hsz - 1 : y);  // scalar
        uint32_t goff = ((uint32_t)(ci * S_SPATIAL) +
                         (uint32_t)(b * DHW + z * HW + y * Wsz + wcl)) * 4u;
        uint32_t loff = (uint32_t)(uintptr_t)(&tile[rr * 132 + xx]);
        asm volatile("global_load_async_to_lds_b32 %0, %1, %2"
                     :: "v"(loff), "v"(goff), "s"(pooled) : "memory");
    }
    asm volatile("s_wait_asynccnt 0" ::: "memory");
    __syncthreads();   // all waves waited their own ASYNCcnt -> tile fully landed

    // ---- zero invalid slots: whole row if z/y OOB (scalar), plus edges ----
#pragma unroll
    for (int rr = 0; rr < 18; ++rr) {
        const int zz = (rr / 3) % 3, yy = rr % 3;
        const bool rowBad = ((unsigned)(d + zz - 1) >= (unsigned)Dsz) ||
                            ((unsigned)(h + yy - 1) >= (unsigned)Hsz);   // scalar
        if (tid < 132 && (rowBad || xx == 0 || xx == 129))
            tile[rr * 132 + xx] = 0.0f;
    }
    __syncthreads();

    // ---- conv as chained WMMA: A = weights (rows identical), B = taps -----
    const int lane = tid & 31;
    const int wv   = tid >> 5;         // 8 waves x 16 outputs
    const int w0   = wv * 16;
    const int n    = lane & 15;        // N column = output w offset
    const int hi   = lane >> 4;        // K-half selector (f32 A/B VGPR layout)
    const int wpos = w0 + n;

    v8f acc = {};
#pragma unroll
    for (int kb = 0; kb < 56; kb += 4) {
        int base0 = hi ? tapBase(kb + 2) : tapBase(kb);       // cndmask of imms
        int base1 = hi ? tapBase(kb + 3) : tapBase(kb + 1);
        v2f a = *(const v2f*)(&wLds[kb + 2 * hi]);            // one ds_load_b64
        v2f bm;
        bm.x = tile[base0 + wpos];
        bm.y = tile[base1 + wpos];
        acc = __builtin_amdgcn_wmma_f32_16x16x4_f32(
            false, a, false, bm, (short)0, acc, false, false);
    }
    // All 16 D rows identical; acc[0] = attn for column n in every lane.
    float attn = acc[0] + cb[0];
    float sg = 1.0f / (1.0f + __expf(-attn));
    if (lane < 16) sAttn[w0 + lane] = sg;
    __syncthreads();

    // ---- dominant phase: NT b128 stream, fully coalesced ------------------
    const long xbase = (long)b * Csz * DHW + (long)d * HW + (long)h * Wsz;
#pragma unroll
    for (int it = 0; it < 4; ++it) {
        int i  = it * 256 + tid;              // 32 channels x 32 quads
        int c  = i >> 5;
        int wq = i & 31;
        long idx = xbase + (long)c * DHW + (long)(wq * 4);
        v4f xv  = __builtin_nontemporal_load((const v4f*)(x + idx));
        v4f at4 = *(const v4f*)(&sAttn[wq * 4]);              // ds_load_b128
        v4f o   = xv * at4;
        __builtin_nontemporal_store(o, (v4f*)(out + idx));
    }
}

// ---------------------------------------------------------------------------
extern "C" void kernel_launch(void* const* d_in, const int* in_sizes, int n_in,
                              void* d_out, int out_size, void* d_ws, size_t ws_size,
                              hipStream_t stream) {
    const float* x  = (const float*)d_in[0];
    const float* cw = (const float*)d_in[1];   // 54 floats [1,2,3,3,3]
    const float* cb = (const float*)d_in[2];   // 1 float
    float* out    = (float*)d_out;
    float* pooled = (float*)d_ws;              // maxP | avgP planes

    pool_kernel<<<dim3(S_SPATIAL / 4 / 256), dim3(256), 0, stream>>>(
        x, pooled, pooled + S_SPATIAL);
    attn_kernel<<<dim3(Bsz * Dsz * Hsz), dim3(256), 0, stream>>>(
        x, cw, cb, pooled, out);
}